// KineticOptimalDiscreteEulerSolver_20658792694014
// MI455X (gfx1250) — compile-verified
//
#include <hip/hip_runtime.h>

#define VV 1024
#define KOEPS 1e-8f
#define KOTINY 1.1754943508222875e-38f   // finfo(float32).tiny

typedef __attribute__((ext_vector_type(2)))  float    v2f;
typedef __attribute__((ext_vector_type(8)))  float    v8f;
typedef int ko_v4i __attribute__((vector_size(16)));            // int __vector(4)
typedef __attribute__((address_space(1))) ko_v4i* ko_gvp;       // global int4*
typedef __attribute__((address_space(3))) ko_v4i* ko_lvp;       // LDS int4*

__device__ __forceinline__ unsigned ko_rotl32(unsigned x, unsigned r) {
    return (x << r) | (x >> (32u - r));
}

// JAX threefry2x32 with key = (0, 1)  (jax.random.key(1))
__device__ __forceinline__ unsigned ko_threefry(unsigned x0, unsigned x1, int which) {
    const unsigned ks0 = 0u, ks1 = 1u, ks2 = 0u ^ 1u ^ 0x1BD11BDAu; // 0x1BD11BDB
    x0 += ks0; x1 += ks1;
#define KO_R(r) { x0 += x1; x1 = ko_rotl32(x1, (r)); x1 ^= x0; }
    KO_R(13) KO_R(15) KO_R(26) KO_R(6)   x0 += ks1; x1 += ks2 + 1u;
    KO_R(17) KO_R(29) KO_R(16) KO_R(24)  x0 += ks2; x1 += ks0 + 2u;
    KO_R(13) KO_R(15) KO_R(26) KO_R(6)   x0 += ks0; x1 += ks1 + 3u;
    KO_R(17) KO_R(29) KO_R(16) KO_R(24)  x0 += ks1; x1 += ks2 + 4u;
    KO_R(13) KO_R(15) KO_R(26) KO_R(6)   x0 += ks2; x1 += ks0 + 5u;
#undef KO_R
    return which ? x1 : x0;
}

// Gumbel noise for flat element e of a 65536-element jax.random.gumbel draw.
// JAX counter layout: e<32768 -> word0 of (e, e+32768); else word1 of (e-32768, e).
__device__ __forceinline__ float ko_gumbel(unsigned e) {
    unsigned bits = (e < 32768u) ? ko_threefry(e, e + 32768u, 0)
                                 : ko_threefry(e - 32768u, e, 1);
    float f = __uint_as_float((bits >> 9) | 0x3f800000u) - 1.0f;  // [0,1)
    float u = fmaxf(KOTINY, f * (1.0f - KOTINY) + KOTINY);        // uniform(tiny, 1)
    return -logf(-logf(u));
}

__device__ __forceinline__ float ko_rcp(float x) {
#if __has_builtin(__builtin_amdgcn_rcpf)
    return __builtin_amdgcn_rcpf(x);   // v_rcp_f32, ~1 ulp
#else
    return 1.0f / x;
#endif
}

__device__ __forceinline__ void ko_wait_async0() {
#if __has_builtin(__builtin_amdgcn_s_wait_asynccnt)
    __builtin_amdgcn_s_wait_asynccnt(0);
#else
    asm volatile("s_wait_asynccnt 0x0" ::: "memory");
#endif
}

__device__ __forceinline__ void ko_wait_store0() {
#if __has_builtin(__builtin_amdgcn_s_wait_storecnt)
    __builtin_amdgcn_s_wait_storecnt(0);
#else
    asm volatile("s_wait_storecnt 0x0" ::: "memory");
#endif
}

// ---- kernel 0: sum(source_p) -> sumbuf -------------------------------------
__global__ void ko_sum(const float* __restrict__ srcp, float* __restrict__ sumbuf) {
    __shared__ float red[256];
    int tid = threadIdx.x;
    float s = 0.0f;
    for (int v = tid; v < VV; v += 256) s += srcp[v];
    red[tid] = s;
    __syncthreads();
    for (int w = 128; w > 0; w >>= 1) {
        if (tid < w) red[tid] += red[tid + w];
        __syncthreads();
    }
    if (tid == 0) *sumbuf = red[0];
}

// ---- kernel 1: x_1[n] = argmax_v(logits[n,v] + gumbel) ---------------------
__global__ void ko_argmax(const float* __restrict__ logits, int* __restrict__ x1buf) {
    __shared__ float rv[256];
    __shared__ int   ri[256];
    int n = blockIdx.x, tid = threadIdx.x;
    float best = -3.402823466e+38f;
    int   bi   = 0;
    for (int v = tid; v < VV; v += 256) {
        float val = logits[n * VV + v] + ko_gumbel((unsigned)(n * VV + v));
        if (val > best) { best = val; bi = v; }
    }
    rv[tid] = best; ri[tid] = bi;
    __syncthreads();
    for (int w = 128; w > 0; w >>= 1) {
        if (tid < w) {
            float ov = rv[tid + w]; int oi = ri[tid + w];
            if (ov > rv[tid] || (ov == rv[tid] && oi < ri[tid])) { rv[tid] = ov; ri[tid] = oi; }
        }
        __syncthreads();
    }
    if (tid == 0) x1buf[n] = ri[0];
}

// ---- kernel 2: rate row per token; row sums via WMMA ones-matvec ------------
// 1 wave per block, 16 tokens per wave (token <-> WMMA column n = lane%16).
__global__ void ko_main(const float* __restrict__ source_p,
                        const int*   __restrict__ x_t,
                        const float* __restrict__ t,
                        const int*   __restrict__ x1buf,
                        const float* __restrict__ sumbuf,
                        float*       __restrict__ out) {
    __shared__ float sp[VV];                      // raw source_p; normalized on use
    int lane = threadIdx.x;                       // 0..31 (wave32)

#if __has_builtin(__builtin_amdgcn_global_load_async_to_lds_b128)
    // gfx1250 async global->LDS copy (ASYNCcnt): 16B/lane, 512B/inst, 8 insts = 4KB
    for (int chunk = 0; chunk < 8; ++chunk) {
        int base = chunk * 128 + lane * 4;        // float index, 16B aligned per lane
        __builtin_amdgcn_global_load_async_to_lds_b128(
            (ko_gvp)(source_p + base),
            (ko_lvp)(&sp[base]),
            0, 0);
    }
    ko_wait_async0();
    __syncthreads();
#else
    for (int v = lane; v < VV; v += 32) sp[v] = source_p[v];
    __syncthreads();
#endif

    float inv = ko_rcp(*sumbuf);                  // 1 / sum(source_p)

    int tl   = lane & 15;
    int half = lane >> 4;
    int tok  = blockIdx.x * 16 + tl;

    float k   = t[0];
    float omk = 1.0f - k;
    int   xt  = x_t[tok];
    int   x1  = x1buf[tok];
    float sxt = sp[xt] * inv;
    float dd  = (xt == x1) ? 1.0f : 0.0f;
    float a   = omk * sxt + k * dd;               // p_t[tok, xt]
    float ad  = dd - sxt;                         // p_t_dot[tok, xt]

    auto uval = [&](int j) -> float {
        float spj = sp[j] * inv;
        float d   = (j == x1) ? 1.0f : 0.0f;
        float pj  = omk * spj + k * d;
        float pdj = d - spj;
        float jt  = fmaxf(a * pdj - ad * pj, 0.0f);
        return jt * ko_rcp(pj + KOEPS);
    };

    v8f c = {};  // D[m,n] accumulates sum_j u[token n, j], replicated over rows m
    v2f ones; ones[0] = 1.0f; ones[1] = 1.0f;     // A = 16x4 all-ones

    for (int jb = 0; jb < VV; jb += 4) {
        int j0 = jb + half;
        int j1 = jb + 2 + half;
        float u0 = uval(j0);
        float u1 = uval(j1);
        // unconditional store; u[xt] == 0 exactly and is overwritten below
        // (same-wave stores to an address are kept in order)
        out[tok * VV + j0] = u0;
        out[tok * VV + j1] = u1;
        v2f b; b[0] = u0; b[1] = u1;              // B = 4x16, column n = lane%16
        c = __builtin_amdgcn_wmma_f32_16x16x4_f32(
                false, ones, false, b, (short)0, c, false, false);
    }

    // diagonal: u[xt] == 0 exactly, so out[tok, xt] = -row_sum.
    // c[0] lanes 0..15 hold D[0, n] = S_token(n); lanes 16..31 duplicate (row 8).
    ko_wait_store0();                             // guarantee overwrite lands last
    if (lane < 16) out[tok * VV + xt] = -c[0];
}

extern "C" void kernel_launch(void* const* d_in, const int* in_sizes, int n_in,
                              void* d_out, int out_size, void* d_ws, size_t ws_size,
                              hipStream_t stream) {
    (void)in_sizes; (void)n_in; (void)out_size; (void)ws_size;
    const float* logits   = (const float*)d_in[0];  // [4,16,1024] f32
    const float* source_p = (const float*)d_in[1];  // [1024] f32
    const int*   x_t      = (const int*)d_in[2];    // [4,16] i32
    const float* t        = (const float*)d_in[3];  // [1] f32
    float* out = (float*)d_out;                     // [4,16,1024] f32

    int*   x1buf  = (int*)d_ws;                     // 64 ints
    float* sumbuf = (float*)d_ws + 64;              // 1 float (byte offset 256)

    ko_sum   <<<1,  256, 0, stream>>>(source_p, sumbuf);
    ko_argmax<<<64, 256, 0, stream>>>(logits, x1buf);
    ko_main  <<<4,  32,  0, stream>>>(source_p, x_t, t, x1buf, sumbuf, out);
}